// TargetMaker_63333587747388
// MI455X (gfx1250) — compile-verified
//
#include <hip/hip_runtime.h>
#include <stdint.h>

#define AS1 __attribute__((address_space(1)))
#define AS3 __attribute__((address_space(3)))

typedef int tm_v4i __attribute__((ext_vector_type(4)));
typedef int tm_v2i __attribute__((ext_vector_type(2)));

#if defined(__HIP_DEVICE_COMPILE__) &&                                   \
    __has_builtin(__builtin_amdgcn_global_load_async_to_lds_b128) &&     \
    __has_builtin(__builtin_amdgcn_global_load_async_to_lds_b64) &&      \
    __has_builtin(__builtin_amdgcn_s_wait_asynccnt)
#define TM_USE_ASYNC_LDS 1
#else
#define TM_USE_ASYNC_LDS 0
#if defined(__HIP_DEVICE_COMPILE__)
#warning "gfx1250 async-to-LDS builtins unavailable; using plain LDS staging fallback"
#endif
#endif

#define TM_N 64  // objects per batch (fixed by reference)

// ---------------------------------------------------------------------------
// Kernel 1: zero the packed (iou,anchor) column-max table in workspace.
// ---------------------------------------------------------------------------
__global__ void tm_init(unsigned long long* __restrict__ p, int n) {
  int i = blockIdx.x * blockDim.x + threadIdx.x;
  if (i < n) p[i] = 0ull;
}

// ---------------------------------------------------------------------------
// Kernel 2: per-object best anchor (argmax over anchors of IoU).
// Thread layout: 256 threads = 4 teams of 64; thread owns object n = tid&63,
// team scans a contiguous anchor chunk (ascending -> first-max tie-break).
// Cross-team merge via atomicMax on packed u64: (f32 bits << 32) | ~anchor,
// monotone for IoU >= 0; ties pick the smallest anchor index (matches jnp).
// ---------------------------------------------------------------------------
__global__ void tm_colmax(const float* __restrict__ gt_boxes,
                          const float* __restrict__ anchors,
                          unsigned long long* __restrict__ best,
                          int A, int chunk) {
  const int b    = blockIdx.y;
  const int n    = threadIdx.x & 63;
  const int team = blockIdx.x * (blockDim.x >> 6) + (threadIdx.x >> 6);
  const int a0   = team * chunk;
  const int a1   = min(a0 + chunk, A);
  if (a0 >= A) return;

  const float* gb = gt_boxes + (size_t)(b * TM_N + n) * 4;
  const float bx0 = gb[0], by0 = gb[1], bx1 = gb[2], by1 = gb[3];
  const float areaB = (bx1 - bx0) * (by1 - by0);

  float    bestV = -1.0f;
  unsigned bestA = 0u;
  for (int a = a0; a < a1; ++a) {
    const float4 c = reinterpret_cast<const float4*>(anchors)[a];
    const float ax0 = c.x - c.z * 0.5f, ay0 = c.y - c.w * 0.5f;
    const float ax1 = c.x + c.z * 0.5f, ay1 = c.y + c.w * 0.5f;
    const float lx = fmaxf(ax0, bx0), ly = fmaxf(ay0, by0);
    const float rx = fminf(ax1, bx1), ry = fminf(ay1, by1);
    const float w = fmaxf(rx - lx, 0.0f), h = fmaxf(ry - ly, 0.0f);
    const float inter = w * h;
    const float areaA = (ax1 - ax0) * (ay1 - ay0);
    const float iou = inter / (areaA + areaB - inter);
    if (iou > bestV) { bestV = iou; bestA = (unsigned)a; }
  }
  const unsigned long long packed =
      ((unsigned long long)__float_as_uint(bestV) << 32) |
      (unsigned long long)(~bestA);
  atomicMax(best + (size_t)b * TM_N + n, packed);
}

// ---------------------------------------------------------------------------
// Kernel 3: per-anchor row argmax with force-match fix + outputs.
// Stages the batch's 64 gt boxes (1KB) and labels (256B) into LDS via the
// CDNA5 async-to-LDS path (wave 0 issues b128/b64 async loads, then
// s_wait_asynccnt 0 before the workgroup barrier).
// ---------------------------------------------------------------------------
__global__ void tm_rows(const float* __restrict__ gt_boxes,
                        const int* __restrict__ gt_labels,
                        const float* __restrict__ anchors,
                        const unsigned long long* __restrict__ best,
                        float* __restrict__ out_classes,
                        float* __restrict__ out_loc,
                        float* __restrict__ out_mask,
                        int A) {
  __shared__ __align__(16) float sBox[TM_N * 4];
  __shared__ __align__(8) int sLabel[TM_N];
  __shared__ int sObj[TM_N];

  const int b   = blockIdx.y;
  const int tid = threadIdx.x;
  const float* gbox = gt_boxes + (size_t)b * TM_N * 4;
  const int*   glab = gt_labels + (size_t)b * TM_N;

#if TM_USE_ASYNC_LDS
  if (tid < 32) {
    // boxes: 1024B = 2 x (32 lanes x 16B); imm offset applies to both sides
    __builtin_amdgcn_global_load_async_to_lds_b128(
        (AS1 tm_v4i*)(gbox + tid * 4), (AS3 tm_v4i*)(sBox + tid * 4), 0, 0);
    __builtin_amdgcn_global_load_async_to_lds_b128(
        (AS1 tm_v4i*)(gbox + tid * 4), (AS3 tm_v4i*)(sBox + tid * 4), 512, 0);
    // labels: 256B = 32 lanes x 8B
    __builtin_amdgcn_global_load_async_to_lds_b64(
        (AS1 tm_v2i*)(glab + tid * 2), (AS3 tm_v2i*)(sLabel + tid * 2), 0, 0);
    __builtin_amdgcn_s_wait_asynccnt(0);
  }
#else
  if (tid < TM_N) {
    reinterpret_cast<float4*>(sBox)[tid] =
        reinterpret_cast<const float4*>(gbox)[tid];
    sLabel[tid] = glab[tid];
  }
#endif
  if (tid < TM_N) {
    // unpack forced-match anchor index (low dword holds ~anchor)
    sObj[tid] = (int)(~(unsigned)(best[(size_t)b * TM_N + tid] & 0xFFFFFFFFull));
  }
  __syncthreads();

  const int a = blockIdx.x * blockDim.x + tid;
  if (a >= A) return;

  const float4 c = reinterpret_cast<const float4*>(anchors)[a];
  const float ax0 = c.x - c.z * 0.5f, ay0 = c.y - c.w * 0.5f;
  const float ax1 = c.x + c.z * 0.5f, ay1 = c.y + c.w * 0.5f;
  const float areaA = (ax1 - ax0) * (ay1 - ay0);

  float bestV = -1.0f;
  int   bestN = 0;
  for (int n = 0; n < TM_N; ++n) {
    const float4 g = reinterpret_cast<const float4*>(sBox)[n];
    const float lx = fmaxf(ax0, g.x), ly = fmaxf(ay0, g.y);
    const float rx = fminf(ax1, g.z), ry = fminf(ay1, g.w);
    const float w = fmaxf(rx - lx, 0.0f), h = fmaxf(ry - ly, 0.0f);
    const float inter = w * h;
    const float areaB = (g.z - g.x) * (g.w - g.y);
    const float iou = inter / (areaA + areaB - inter);
    if (iou > bestV) { bestV = iou; bestN = n; }  // strict > == first-max
  }
  // Force-match: entries (obj_idx[n], n) become exactly 1.0, then argmax with
  // first-occurrence semantics. Ascending scan reproduces that exactly.
  for (int n = 0; n < TM_N; ++n) {
    if (sObj[n] == a) {
      if (1.0f > bestV || (bestV == 1.0f && n < bestN)) { bestV = 1.0f; bestN = n; }
    }
  }

  const bool pos = bestV >= 0.5f;
  const float4 g = reinterpret_cast<const float4*>(sBox)[bestN];
  // xy -> cxcy (division by 2 is exact as *0.5f)
  const float gcx = (g.x + g.z) * 0.5f;
  const float gcy = (g.y + g.w) * 0.5f;
  const float gw  = g.z - g.x;
  const float gh  = g.w - g.y;
  // SSD variance encode, matching reference expression order:
  //   (g_cxcy - a_cxcy) / (a_wh / 10.0) ; log(g_wh / a_wh) * 5.0
  const float l0 = (gcx - c.x) / (c.z / 10.0f);
  const float l1 = (gcy - c.y) / (c.w / 10.0f);
  const float l2 = logf(gw / c.z) * 5.0f;
  const float l3 = logf(gh / c.w) * 5.0f;

  const size_t idx = (size_t)b * A + a;
  out_classes[idx] = (float)((sLabel[bestN] + 1) * (pos ? 1 : 0));
  float* lp = out_loc + idx * 4;
  lp[0] = l0; lp[1] = l1; lp[2] = l2; lp[3] = l3;
  out_mask[idx] = pos ? 1.0f : 0.0f;
}

// ---------------------------------------------------------------------------
extern "C" void kernel_launch(void* const* d_in, const int* in_sizes, int n_in,
                              void* d_out, int out_size, void* d_ws, size_t ws_size,
                              hipStream_t stream) {
  (void)n_in; (void)out_size; (void)ws_size;
  const float* gt_boxes  = (const float*)d_in[0];   // [B,N,4] f32
  const int*   gt_labels = (const int*)d_in[1];     // [B,N]   i32
  const float* anchors   = (const float*)d_in[2];   // [A,4]   f32 (cxcywh)

  const int BN = in_sizes[1];        // B*N
  const int B  = BN / TM_N;          // N fixed at 64 by the reference
  const int A  = in_sizes[2] / 4;

  unsigned long long* best = (unsigned long long*)d_ws;   // B*N packed maxima (32KB)

  float* out_classes = (float*)d_out;
  float* out_loc     = out_classes + (size_t)B * A;
  float* out_mask    = out_loc + (size_t)B * A * 4;

  // Pass 0: init column-max table
  tm_init<<<(BN + 255) / 256, 256, 0, stream>>>(best, BN);

  // Pass 1: best anchor per object (96 teams of 64 per batch)
  const int gridX = 24;                        // 24 blocks * 4 teams = 96 teams
  const int teams = gridX * 4;
  const int chunk = (A + teams - 1) / teams;   // 256 anchors per team for A=24564
  tm_colmax<<<dim3(gridX, B), 256, 0, stream>>>(gt_boxes, anchors, best, A, chunk);

  // Pass 2: per-anchor matching + outputs
  tm_rows<<<dim3((A + 255) / 256, B), 256, 0, stream>>>(
      gt_boxes, gt_labels, anchors, best, out_classes, out_loc, out_mask, A);
}